// D3PM_19894288515513
// MI455X (gfx1250) — compile-verified
//
#include <hip/hip_runtime.h>
#include <hip/hip_bf16.h>
#include <stdint.h>

// D3PM q_sample for structure (N=4100, mask=3) and sequence (N=25, mask=24).
// B=16, L=512. Output: [noised_structure(8192) | noised_seq(8192) | t(16)] as float.

#define B_DIM 16
#define L_DIM 512
#define NTOK (B_DIM * L_DIM)   // 8192
#define MASK_STRUC 3
#define MASK_SEQ 24
#define TILE 256               // tokens per block

typedef __attribute__((ext_vector_type(4))) unsigned u32x4;
typedef __attribute__((ext_vector_type(8))) unsigned u32x8;

// Deterministic counter-based RNG (splitmix/murmur finalizer) — stateless,
// identical across graph replays.
__device__ __forceinline__ unsigned mix32(unsigned long long x) {
    x ^= x >> 33; x *= 0xff51afd7ed558ccdULL;
    x ^= x >> 33; x *= 0xc4ceb9fe1a85ec53ULL;
    x ^= x >> 33;
    return (unsigned)x;
}

// Issue one TDM tensor_load_to_lds moving a contiguous 256 x int32 tile
// (1024 bytes) from global memory into LDS at byte offset lds_off.
// D# layout per CDNA5 ISA:
//   group0: [0]=count=1, [1]=lds_addr, [2]=global_addr[31:0],
//           [3]=global_addr[56:32] | type(2)<<30
//   group1: [0]=data_size(4B)=2<<16, tensor_dim0=256 (bits 79:48),
//           tensor_dim1=1 (bits 111:80), tile_dim0=256 (bits 127:112),
//           tile_dim1=1, tile_dim2=0, tensor_dim0_stride=256, dim1_stride=0
//   groups 2/3: zero (2D tensor).
__device__ __forceinline__ void tdm_load_tile(unsigned lds_off, const int* gptr) {
    unsigned long long ga = (unsigned long long)(uintptr_t)gptr;
    u32x4 g0 = { 1u,
                 lds_off,
                 (unsigned)ga,
                 ((unsigned)(ga >> 32) & 0x01FFFFFFu) | (2u << 30) };
    u32x8 g1 = { 0x00020000u,          // data_size = 2 (4 bytes), wg_mask = 0
                 256u << 16,           // tensor_dim0[15:0] at bits 63:48
                 1u << 16,             // tensor_dim0 hi = 0, tensor_dim1 = 1
                 256u << 16,           // tensor_dim1 hi = 0, tile_dim0 = 256
                 1u,                   // tile_dim1 = 1, tile_dim2 = 0
                 256u,                 // tensor_dim0_stride = 256
                 0u,                   // stride hi, tensor_dim1_stride lo
                 0u };                 // tensor_dim1_stride hi
    u32x4 gz = { 0u, 0u, 0u, 0u };
    asm volatile("tensor_load_to_lds %0, %1, %2, %3"
                 :: "s"(g0), "s"(g1), "s"(gz), "s"(gz)
                 : "memory");
}

__global__ __launch_bounds__(TILE) void D3PM_19894288515513_kernel(
        const int* __restrict__ structure,
        const int* __restrict__ sequence,
        const int* __restrict__ t,
        const float* __restrict__ alpha,
        float* __restrict__ out) {
    __shared__ int s_struct[TILE];
    __shared__ int s_seq[TILE];

    const int tid  = threadIdx.x;
    const int base = blockIdx.x * TILE;

    // Wave 0 stages this block's token tiles into LDS via the Tensor Data
    // Mover (async DMA, tracked by TENSORcnt), then drains the counter.
    if (tid < 32) {
        if (tid == 0) {
            __builtin_prefetch(alpha, 0, 0);   // global_prefetch_b8 for alpha table
        }
        tdm_load_tile((unsigned)(uintptr_t)s_struct, structure + base);
        tdm_load_tile((unsigned)(uintptr_t)s_seq,    sequence  + base);
        __builtin_amdgcn_s_wait_tensorcnt(0);
    }
    __syncthreads();   // publish LDS tiles to the other 7 waves

    const int   i    = base + tid;        // 0..8191
    const int   b    = i >> 9;            // batch index (L=512)
    const int   tval = t[b];
    const float a    = alpha[tval];       // keep probability

    // Two independent uniforms per token (structure / sequence streams).
    const float us = (float)(mix32(((unsigned long long)i << 1) ^ 0x9E3779B97F4A7C15ULL) >> 8)
                     * (1.0f / 16777216.0f);
    const float uq = (float)(mix32(((unsigned long long)i << 1) + 0xD1B54A32D192ED03ULL) >> 8)
                     * (1.0f / 16777216.0f);

    const int x0s = s_struct[tid];
    const int x0q = s_seq[tid];

    // Two-point categorical: keep x0 w.p. a, else absorb into mask token.
    const int toks = (us < a) ? x0s : MASK_STRUC;
    const int tokq = (uq < a) ? x0q : MASK_SEQ;

    out[i]        = (float)toks;
    out[NTOK + i] = (float)tokq;
    if (i < B_DIM) {
        out[2 * NTOK + i] = (float)t[i];  // pass-through t
    }
}

extern "C" void kernel_launch(void* const* d_in, const int* in_sizes, int n_in,
                              void* d_out, int out_size, void* d_ws, size_t ws_size,
                              hipStream_t stream) {
    const int*   structure = (const int*)d_in[0];   // (16, 512) int32
    const int*   sequence  = (const int*)d_in[1];   // (16, 512) int32
    const int*   t         = (const int*)d_in[2];   // (16,) int32
    const float* alpha     = (const float*)d_in[3]; // (1001,) float32
    float* out = (float*)d_out;                     // 16400 floats

    D3PM_19894288515513_kernel<<<NTOK / TILE, TILE, 0, stream>>>(
        structure, sequence, t, alpha, out);
}